// Net_75230647156970
// MI455X (gfx1250) — compile-verified
//
#include <hip/hip_runtime.h>
#include <cstddef>

// ---------------------------------------------------------------------------
// MoE micro-MLP, 16 samples per wave, all GEMMs via V_WMMA_F32_16X16X4_F32.
// Intermediates kept transposed (features x samples) in WMMA C/D layout:
//   lane l (<16):  VGPR i -> (feat = i,      sample = l)
//   lane l (>=16): VGPR i -> (feat = i + 8,  sample = l - 16)
// Biases are applied through the matrix pipe via K-augmentation: one extra
// WMMA per GEMM with A = bias column of [W^T | b] and B = shared ones-row
// fragment. This removes all bias C-init register tiles (VGPR pressure was
// pushing past 256 and forcing s_set_vgpr_msb churn).
// All transcendentals are branch-free (native v_exp/v_log/v_rcp/v_rsq).
// ---------------------------------------------------------------------------

typedef __attribute__((ext_vector_type(2))) float v2f;
typedef __attribute__((ext_vector_type(8))) float v8f;

#define LN_EPS 1e-5f

// Exchange value with lane ^ 16 (SWAPX16 via ds_swizzle: xor=0x10, and=0x1f).
__device__ __forceinline__ float swap16(float v) {
  return __int_as_float(__builtin_amdgcn_ds_swizzle(__float_as_int(v), 0x401F));
}

__device__ __forceinline__ v8f wmma4(v2f a, v2f b, v8f c) {
  // D = A(16x4) * B(4x16) + C, fp32.
  return __builtin_amdgcn_wmma_f32_16x16x4_f32(false, a, false, b, (short)0, c,
                                               false, false);
}

__device__ __forceinline__ v8f zeroTile() {
  v8f z;
#pragma unroll
  for (int i = 0; i < 8; ++i) z[i] = 0.f;
  return z;
}

// ---- branch-free fast math (single v_exp/v_log/v_rcp instructions) --------
__device__ __forceinline__ float rcpf(float v) {
  return __builtin_amdgcn_rcpf(v);
}
__device__ __forceinline__ float fast_tanh(float v) {
  // 1 - 2/(e^{2v}+1); exp->inf gives 1, exp->0 gives -1. No branches.
  return 1.f - 2.f * rcpf(1.f + __expf(2.f * v));
}
__device__ __forceinline__ float siluf(float v) {
  return v * rcpf(1.f + __expf(-v));
}
__device__ __forceinline__ float softplusf(float v) {
  return fmaxf(v, 0.f) + __logf(1.f + __expf(-fabsf(v)));
}

// Build a B-fragment (K-chunk KB..KB+3, N = samples) from a D-layout tile.
// B layout: VGPR0 = row K=KB (lanes<16) / K=KB+2 (lanes>=16),
//           VGPR1 = row K=KB+1 / K=KB+3.
template <int KB>
__device__ __forceinline__ v2f bfrag(v8f t, int hi) {
  constexpr int b = KB & 7;
  float lx = t[b], ly = t[b + 1];
  float hx = t[b + 2], hy = t[b + 3];
  v2f r;
  if (KB < 8) {
    // rows KB..KB+3 live in the low-half lanes of the tile
    float cx = swap16(hx), cy = swap16(hy);
    r.x = hi ? cx : lx;
    r.y = hi ? cy : ly;
  } else {
    // rows KB..KB+3 live in the high-half lanes of the tile
    float cx = swap16(lx), cy = swap16(ly);
    r.x = hi ? hx : cx;
    r.y = hi ? hy : cy;
  }
  return r;
}

// A-fragment of W^T (W stored row-major (in_dim, out_dim)):
// lane m = mbase + (lane&15), K pair = kb + 2*hi. Rows >= mvalid are zero.
__device__ __forceinline__ v2f aFrag(const float* W, int out_dim, int mbase,
                                     int kb, int mvalid) {
  int lane = threadIdx.x & 31;
  int hi = lane >> 4;
  int m = mbase + (lane & 15);
  int k = kb + (hi ? 2 : 0);
  v2f a;
  a.x = 0.f;
  a.y = 0.f;
  if (m < mvalid) {
    a.x = W[k * out_dim + m];
    a.y = W[(k + 1) * out_dim + m];
  }
  return a;
}

// Bias A-fragment: augmentation chunk. Row K=+0 (low-half lanes only) holds
// the bias column b[m]; all other rows of the chunk are zero.
__device__ __forceinline__ v2f biasFrag(const float* b, int mbase,
                                        int mvalid) {
  int lane = threadIdx.x & 31;
  int hi = lane >> 4;
  int m = mbase + (lane & 15);
  v2f a;
  a.x = (!hi && m < mvalid) ? b[m] : 0.f;
  a.y = 0.f;
  return a;
}

// GEMM with X already in B-fragment form (K = 16).
__device__ __forceinline__ v8f gemmX(const v2f* A, const v2f* xb, v8f c) {
#pragma unroll
  for (int k = 0; k < 4; ++k) c = wmma4(A[k], xb[k], c);
  return c;
}

// GEMM with a D-layout tile as RHS (K = 16), repacking via swizzle.
__device__ __forceinline__ v8f gemmT(const v2f* A, v8f t, int hi, v8f c) {
  c = wmma4(A[0], bfrag<0>(t, hi), c);
  c = wmma4(A[1], bfrag<4>(t, hi), c);
  c = wmma4(A[2], bfrag<8>(t, hi), c);
  c = wmma4(A[3], bfrag<12>(t, hi), c);
  return c;
}

// One expert: h = LN(x@w1+b1)*g+bn -> silu -> silu(h@w2+b2). 16 samples.
__device__ __forceinline__ v8f expertEval(const v2f A0[4], const v2f A1[4],
                                          const v2f B[6], v2f bf0, v2f bf1,
                                          v2f bf2, v2f oneb, const float* g0,
                                          const float* g1, const float* b0,
                                          const float* b1, const v2f xb[4],
                                          int hi) {
  // Bias via augmented-K WMMA, then the K=16 data chunks.
  v8f t0 = wmma4(bf0, oneb, zeroTile());
  v8f t1 = wmma4(bf1, oneb, zeroTile());
#pragma unroll
  for (int k = 0; k < 4; ++k) {
    t0 = wmma4(A0[k], xb[k], t0);
    t1 = wmma4(A1[k], xb[k], t1);
  }
  // LayerNorm over the 24 features of each sample (rows 24..31 are exact 0).
  float sum = 0.f, sq = 0.f;
#pragma unroll
  for (int i = 0; i < 8; ++i) {
    sum += t0[i] + t1[i];
    sq += t0[i] * t0[i] + t1[i] * t1[i];
  }
  sum += swap16(sum);
  sq += swap16(sq);
  float mu = sum * (1.f / 24.f);
  float var = sq * (1.f / 24.f) - mu * mu;
  float rs = __builtin_amdgcn_rsqf(var + LN_EPS);
#pragma unroll
  for (int i = 0; i < 8; ++i) {
    t0[i] = siluf((t0[i] - mu) * rs * g0[i] + b0[i]);
    t1[i] = siluf((t1[i] - mu) * rs * g1[i] + b1[i]);
  }
  // h(24) @ w2(24x16) + b2, K = 24 (+ bias chunk).
  v8f o = wmma4(bf2, oneb, zeroTile());
  o = wmma4(B[0], bfrag<0>(t0, hi), o);
  o = wmma4(B[1], bfrag<4>(t0, hi), o);
  o = wmma4(B[2], bfrag<8>(t0, hi), o);
  o = wmma4(B[3], bfrag<12>(t0, hi), o);
  o = wmma4(B[4], bfrag<0>(t1, hi), o);
  o = wmma4(B[5], bfrag<4>(t1, hi), o);
#pragma unroll
  for (int i = 0; i < 8; ++i) o[i] = siluf(o[i]);
  return o;
}

__global__ __launch_bounds__(256) void moe_heads_kernel(
    const float* __restrict__ x, const float* __restrict__ gw1,
    const float* __restrict__ gb1, const float* __restrict__ gw2,
    const float* __restrict__ gb2, const float* __restrict__ e1w1,
    const float* __restrict__ e1b1, const float* __restrict__ e1g,
    const float* __restrict__ e1bn, const float* __restrict__ e1w2,
    const float* __restrict__ e1b2, const float* __restrict__ e2w1,
    const float* __restrict__ e2b1, const float* __restrict__ e2g,
    const float* __restrict__ e2bn, const float* __restrict__ e2w2,
    const float* __restrict__ e2b2, const float* __restrict__ tw,
    const float* __restrict__ tb, const float* __restrict__ sw,
    const float* __restrict__ sb, const float* __restrict__ hw,
    const float* __restrict__ hb, float* __restrict__ out, int nrows) {
  const int lane = threadIdx.x & 31;
  const int hi = lane >> 4;
  const int l15 = lane & 15;

  // Shared ones-row B fragment for all bias-augmentation WMMAs:
  // row K=+0 (low-half lanes) is 1, all other rows 0.
  v2f oneb;
  oneb.x = hi ? 0.f : 1.f;
  oneb.y = 0.f;

  // ---------------- preload all weight A-fragments (W^T tiles) -------------
  v2f GW1[4], GW2[2], E1A0[4], E1A1[4], E1B[6], E2A0[4], E2A1[4], E2B[6],
      TWF[4], HDF[4];
#pragma unroll
  for (int k = 0; k < 4; ++k) {
    GW1[k] = aFrag(gw1, 8, 0, 4 * k, 8);
    E1A0[k] = aFrag(e1w1, 24, 0, 4 * k, 24);
    E1A1[k] = aFrag(e1w1, 24, 16, 4 * k, 24);
    E2A0[k] = aFrag(e2w1, 24, 0, 4 * k, 24);
    E2A1[k] = aFrag(e2w1, 24, 16, 4 * k, 24);
    TWF[k] = aFrag(tw, 16, 0, 4 * k, 16);
  }
#pragma unroll
  for (int k = 0; k < 6; ++k) {
    E1B[k] = aFrag(e1w2, 16, 0, 4 * k, 16);
    E2B[k] = aFrag(e2w2, 16, 0, 4 * k, 16);
  }
  GW2[0] = aFrag(gw2, 2, 0, 0, 2);
  GW2[1] = aFrag(gw2, 2, 0, 4, 2);
  // Head A-frags: P^T (3x16) with row0 = sw^T, rows 1..2 = hw^T.
#pragma unroll
  for (int c = 0; c < 4; ++c) {
    int k = 4 * c + (hi ? 2 : 0);
    int m = l15;
    v2f a;
    a.x = 0.f;
    a.y = 0.f;
    if (m == 0) {
      a.x = sw[k];
      a.y = sw[k + 1];
    } else if (m < 3) {
      a.x = hw[k * 2 + (m - 1)];
      a.y = hw[(k + 1) * 2 + (m - 1)];
    }
    HDF[c] = a;
  }

  // ---------------- bias A-fragments (augmented-K columns) -----------------
  v2f GB1f = biasFrag(gb1, 0, 8);
  v2f GB2f = biasFrag(gb2, 0, 2);
  v2f E1F0 = biasFrag(e1b1, 0, 24), E1F1 = biasFrag(e1b1, 16, 24);
  v2f E1F2 = biasFrag(e1b2, 0, 16);
  v2f E2F0 = biasFrag(e2b1, 0, 24), E2F1 = biasFrag(e2b1, 16, 24);
  v2f E2F2 = biasFrag(e2b2, 0, 16);
  v2f TBf = biasFrag(tb, 0, 16);
  v2f HDf;  // head bias column: [sb, hb0, hb1, 0...]
  {
    float bv = 0.f;
    if (l15 == 0) bv = sb[0];
    if (l15 == 1) bv = hb[0];
    if (l15 == 2) bv = hb[1];
    HDf.x = hi ? 0.f : bv;
    HDf.y = 0.f;
  }

  // ---------------- LayerNorm gamma/beta per-lane tables -------------------
  float G1a[8], B1a[8], G1b[8], B1b[8], G2a[8], B2a[8], G2b[8], B2b[8];
#pragma unroll
  for (int i = 0; i < 8; ++i) {
    int f0 = i + 8 * hi;
    int f1 = 16 + i + 8 * hi;
    bool v1 = (f1 < 24);
    G1a[i] = e1g[f0];
    B1a[i] = e1bn[f0];
    G1b[i] = v1 ? e1g[f1] : 0.f;
    B1b[i] = v1 ? e1bn[f1] : 0.f;
    G2a[i] = e2g[f0];
    B2a[i] = e2bn[f0];
    G2b[i] = v1 ? e2g[f1] : 0.f;
    B2b[i] = v1 ? e2bn[f1] : 0.f;
  }

  // ---------------- tile loop ----------------------------------------------
  const int tiles = (nrows + 15) >> 4;
  const int nwaves = (int)((gridDim.x * blockDim.x) >> 5);
  int gwave = (int)((blockIdx.x * blockDim.x + threadIdx.x) >> 5);

  for (int t = gwave; t < tiles; t += nwaves) {
    const int row0 = t << 4;
    int r = row0 + l15;
    if (r >= nrows) r = nrows - 1;

    // X^T B-fragments: lane -> sample l15, feature pair (4c + 2*hi).
    const v2f* xr = (const v2f*)(x + (size_t)r * 16);
    v2f xb[4];
#pragma unroll
    for (int c = 0; c < 4; ++c) xb[c] = xr[2 * c + hi];

    // Prefetch next tile of x (global_prefetch_b8).
    int tn = t + nwaves;
    if (tn < tiles)
      __builtin_prefetch(x + ((size_t)(tn << 4) + l15) * 16, 0, 0);

    // ---- gate: softmax(tanh(x@gw1+gb1)@gw2+gb2) ----
    v8f g = wmma4(GB1f, oneb, zeroTile());
    g = gemmX(GW1, xb, g);
#pragma unroll
    for (int i = 0; i < 8; ++i) g[i] = fast_tanh(g[i]);
    v8f lg = wmma4(GB2f, oneb, zeroTile());
    lg = wmma4(GW2[0], bfrag<0>(g, hi), lg);
    lg = wmma4(GW2[1], bfrag<4>(g, hi), lg);
    float l0 = lg[0], l1 = lg[1];
    float mx = fmaxf(l0, l1);
    float ew0 = __expf(l0 - mx), ew1 = __expf(l1 - mx);
    float sden = rcpf(ew0 + ew1);
    float w0 = ew0 * sden;
    float w1 = ew1 * sden;
    {  // broadcast gate weights from low-half lanes to high-half lanes
      float c0 = swap16(w0), c1 = swap16(w1);
      if (hi) {
        w0 = c0;
        w1 = c1;
      }
    }

    // ---- experts ----
    v8f o1 = expertEval(E1A0, E1A1, E1B, E1F0, E1F1, E1F2, oneb, G1a, G1b,
                        B1a, B1b, xb, hi);
    v8f o2 = expertEval(E2A0, E2A1, E2B, E2F0, E2F1, E2F2, oneb, G2a, G2b,
                        B2a, B2b, xb, hi);

    // ---- combine + trunk ----
    v8f h;
#pragma unroll
    for (int i = 0; i < 8; ++i) h[i] = w0 * o1[i] + w1 * o2[i];
    v8f tr = wmma4(TBf, oneb, zeroTile());
    tr = gemmT(TWF, h, hi, tr);
#pragma unroll
    for (int i = 0; i < 8; ++i) tr[i] = siluf(tr[i]);

    // ---- heads (3x16) ----
    v8f S = wmma4(HDf, oneb, zeroTile());
    S = gemmT(HDF, tr, hi, S);

    float strain = S[0];
    float tens = S[1];
    float gap = softplusf(S[2]);
    if (!hi && (row0 + l15) < nrows) {
      size_t o = (size_t)(row0 + l15) * 3;
      out[o + 0] = strain;
      out[o + 1] = tens;
      out[o + 2] = tens - gap;
    }
  }
}

extern "C" void kernel_launch(void* const* d_in, const int* in_sizes, int n_in,
                              void* d_out, int out_size, void* d_ws,
                              size_t ws_size, hipStream_t stream) {
  (void)n_in;
  (void)d_ws;
  (void)ws_size;
  (void)out_size;
  const float* x = (const float*)d_in[0];
  const float* gw1 = (const float*)d_in[1];
  const float* gb1 = (const float*)d_in[2];
  const float* gw2 = (const float*)d_in[3];
  const float* gb2 = (const float*)d_in[4];
  const float* e1w1 = (const float*)d_in[5];
  const float* e1b1 = (const float*)d_in[6];
  const float* e1g = (const float*)d_in[7];
  const float* e1bn = (const float*)d_in[8];
  const float* e1w2 = (const float*)d_in[9];
  const float* e1b2 = (const float*)d_in[10];
  const float* e2w1 = (const float*)d_in[11];
  const float* e2b1 = (const float*)d_in[12];
  const float* e2g = (const float*)d_in[13];
  const float* e2bn = (const float*)d_in[14];
  const float* e2w2 = (const float*)d_in[15];
  const float* e2b2 = (const float*)d_in[16];
  const float* tw = (const float*)d_in[17];
  const float* tb = (const float*)d_in[18];
  const float* sw = (const float*)d_in[19];
  const float* sb = (const float*)d_in[20];
  const float* hw = (const float*)d_in[21];
  const float* hb = (const float*)d_in[22];

  int nrows = in_sizes[0] / 16;
  int tiles = (nrows + 15) / 16;
  const int threads = 256;  // 8 waves per block
  int wpb = threads / 32;
  int blocks = (tiles + wpb - 1) / wpb;
  if (blocks > 4096) blocks = 4096;  // grid-stride: ~4 tiles/wave, amortizes
                                     // the per-wave weight-fragment preload
  if (blocks < 1) blocks = 1;

  moe_heads_kernel<<<blocks, threads, 0, stream>>>(
      x, gw1, gb1, gw2, gb2, e1w1, e1b1, e1g, e1bn, e1w2, e1b2, e2w1, e2b1,
      e2g, e2bn, e2w2, e2b2, tw, tb, sw, sb, hw, hb, (float*)d_out, nrows);
}